// MCPBRNN_Generic_PETconstraint_MIoutputloss_Sigmoid_53695681134754
// MI455X (gfx1250) — compile-verified
//
#include <hip/hip_runtime.h>
#include <cstdint>

// ---------------------------------------------------------------------------
// MCPBRNN scalar-gate scan (hidden_size == 1): one serial nonlinear recurrence
// of length B*T = 262144.  Latency-bound on the carry chain; CDNA5 async
// global->LDS staging (ASYNCcnt) keeps the scanning lane fed from LDS with no
// global-memory latency on the critical path.  Gate constants are pre-scaled
// by -log2(e) so each sigmoid is exactly: fma -> v_exp2 -> add -> v_rcp -> mul.
// ---------------------------------------------------------------------------

#define TSEQ 1024                     // T (time steps per batch row)
#define ROWS_PER_CHUNK 2
#define CH (ROWS_PER_CHUNK * TSEQ)    // 2048 scan steps per LDS chunk (16 KB)
#define NTHREADS 256
#define PASSES (CH / NTHREADS)        // 8 async-load instructions per thread/chunk

#define NEG_LOG2E (-1.4426950408889634f)

// Stage CH float2 steps from global into LDS asynchronously (ASYNCcnt++ x PASSES
// per wave).  GV mode: VDST = per-lane LDS byte address, VADDR = 64-bit address.
__device__ __forceinline__ void async_copy_chunk(const float2* __restrict__ src,
                                                 const float2* dst_lds, int tid) {
  // Low 32 bits of a generic pointer into LDS are the LDS byte address.
  unsigned lbase = (unsigned)(uintptr_t)dst_lds;
#pragma unroll
  for (int p = 0; p < PASSES; ++p) {
    unsigned e  = (unsigned)(p * NTHREADS + tid);
    unsigned la = lbase + e * 8u;
    unsigned long long ga = (unsigned long long)(uintptr_t)(src + e);
    asm volatile("global_load_async_to_lds_b64 %0, %1, off"
                 :: "v"(la), "v"(ga) : "memory");
  }
}

// Scan one batch row of TSEQ steps out of LDS.  M = (b >= time_lag), hoisted.
// m0*/m1*/m2* are the gate affine coefficients pre-scaled by -log2(e).
// Carry: c0 (cell), hxp1 (lagged cell).  Writes the row's 10 outputs.
template <bool M>
__device__ __forceinline__ void scan_row(const float2* __restrict__ lbuf, int b,
                                         float& c0, float& hxp1,
                                         float m0o, float m1o, float m2o,
                                         float m0l, float m1l, float m2l,
                                         float oo1, float ol1,
                                         float* __restrict__ out, int Bn) {
  float so = 0.0f, sl = 0.0f, olc = 0.0f, lc0 = 0.0f;
#pragma unroll 8
  for (int t = 0; t < TSEQ; ++t) {
    float2 u = lbuf[t];                    // ds_load_b64, issued ahead by unroll
    float c0p1 = M ? hxp1 : 0.0f;
    // exp2 argument of -oo2*log2e / -ol2*log2e directly via pre-scaled FMAs.
    float ao = fmaf(c0, m1o, fmaf(c0p1, m2o, m0o));
    float al = fmaf(c0, m1l, fmaf(u.y, m2l, m0l));
    so = oo1 * __builtin_amdgcn_rcpf(1.0f + __builtin_amdgcn_exp2f(ao));
    sl = ol1 * __builtin_amdgcn_rcpf(1.0f + __builtin_amdgcn_exp2f(al));
    // ol - relu(ol - u2/c0_safe) == min(ol, u2/c0) when c0 > 0.  rcp(c0) runs
    // in parallel with the sigmoid chains; its value is only consumed when
    // c0 > 0, so no csafe select is needed.
    bool  pos = c0 > 0.0f;
    float q   = u.y * __builtin_amdgcn_rcpf(c0);
    olc = pos ? fminf(sl, q) : sl;
    lc0 = c0;                              // pre-update cell (for outputs)
    // c1 = (1 - oo - ol_c)*c0 + u1
    float cn = fmaf(-so, c0, fmaf(-olc, c0, c0 + u.x));
    if (M) hxp1 = c0;
    c0 = cn;
  }
  // Only the final step of each row survives (torch overwrites row b every t).
  out[0 * Bn + b] = so * lc0;          // h_n
  out[1 * Bn + b] = lc0;               // c_n
  out[2 * Bn + b] = sl * lc0;          // l_n
  out[3 * Bn + b] = olc * lc0;         // lc_n
  out[4 * Bn + b] = 0.0f;              // bp_n
  out[5 * Bn + b] = 0.0f;              // Gate_ib
  out[6 * Bn + b] = so;                // oo
  out[7 * Bn + b] = sl;                // ol
  out[8 * Bn + b] = olc;               // ol_c
  out[9 * Bn + b] = 1.0f - so - olc;   // f
}

__global__ __launch_bounds__(NTHREADS)
void mcpbrnn_scan(const float* __restrict__ x,
                  const int*   __restrict__ time_lag_p,
                  const float* __restrict__ cmean_p,
                  const float* __restrict__ cstd_p,
                  const float* __restrict__ wr_yom,
                  const float* __restrict__ wr_ylm,
                  const float* __restrict__ wr_yfm,
                  const float* __restrict__ b0_yom,
                  const float* __restrict__ w1_yom,
                  const float* __restrict__ w1_ylm,
                  const float* __restrict__ b0_ylm,
                  const float* __restrict__ w2_yom,
                  const float* __restrict__ w2_ylm,
                  float* __restrict__ out, int Bn, int nsteps) {
  __shared__ float2 buf[2][CH];          // 32 KB double buffer (of 320 KB LDS)
  const int tid = threadIdx.x;
  const int nch = nsteps / CH;
  const float2* xv = reinterpret_cast<const float2*>(x);

  // Prologue: stage chunk 0.
  async_copy_chunk(xv, &buf[0][0], tid);

  // Fold every loop-invariant gate constant into FMA-form coefficients,
  // pre-scaled by -log2(e) so sigmoid needs no extra multiply on the chain.
  const float MLc = 2.9086f, SLc = 1.898f;
  const float cm  = cmean_p[0];
  const float inv = 1.0f / cstd_p[0];
  const float eo = expf(wr_yom[0]), el = expf(wr_ylm[0]), ef = expf(wr_yfm[0]);
  const float rden = 1.0f / (eo + el + ef);
  const float oo1 = eo * rden, ol1 = el * rden;
  const float k1o = inv * w1_yom[0];
  const float k2o = inv * w2_yom[0];
  const float k0o = b0_yom[0] - cm * (k1o + k2o);
  const float k1l = inv * w1_ylm[0];
  const float k2l = w2_ylm[0] / SLc;
  const float k0l = b0_ylm[0] - cm * k1l - (MLc / SLc) * w2_ylm[0];
  const float m0o = NEG_LOG2E * k0o, m1o = NEG_LOG2E * k1o, m2o = NEG_LOG2E * k2o;
  const float m0l = NEG_LOG2E * k0l, m1l = NEG_LOG2E * k1l, m2l = NEG_LOG2E * k2l;
  const int   tl  = time_lag_p[0];

  float c0 = 0.0f, hxp1 = 0.0f;          // scan carry (lives in lane 0's regs)

  for (int c = 0; c < nch; ++c) {
    if (c + 1 < nch) {
      // Issue next chunk (its buffer was last read 2 chunks ago; the trailing
      // barrier of the previous iteration ordered that read).
      async_copy_chunk(xv + (size_t)(c + 1) * CH, &buf[(c + 1) & 1][0], tid);
      // Keep the PASSES just-issued ops in flight; in-order completion means
      // ASYNCcnt <= PASSES implies chunk c's data is resident in LDS.
      asm volatile("s_wait_asynccnt 0x8" ::: "memory");
    } else {
      asm volatile("s_wait_asynccnt 0x0" ::: "memory");
    }
    __syncthreads();                     // all waves' slices of chunk c visible

    if (tid == 0) {
      const float2* lb = &buf[c & 1][0];
#pragma unroll
      for (int r = 0; r < ROWS_PER_CHUNK; ++r) {
        int b = c * ROWS_PER_CHUNK + r;
        if (b >= tl)
          scan_row<true >(lb + r * TSEQ, b, c0, hxp1,
                          m0o, m1o, m2o, m0l, m1l, m2l, oo1, ol1, out, Bn);
        else
          scan_row<false>(lb + r * TSEQ, b, c0, hxp1,
                          m0o, m1o, m2o, m0l, m1l, m2l, oo1, ol1, out, Bn);
      }
    }
    __syncthreads();                     // scan done before buffer is reused
  }
}

// ---------------------------------------------------------------------------
// Host side
// ---------------------------------------------------------------------------
extern "C" void kernel_launch(void* const* d_in, const int* in_sizes, int n_in,
                              void* d_out, int out_size, void* d_ws, size_t ws_size,
                              hipStream_t stream) {
  // setup_inputs() order:
  // 0:x 1:epoch 2:time_lag 3:y_obs 4:cmean 5:cstd 6:weight_r_yom 7:weight_r_ylm
  // 8:weight_r_yfm 9:bias_b0_yom 10:weight_b1_yom 11:weight_b1_ylm
  // 12:bias_b0_ylm 13:weight_b2_yom 14:weight_b2_ylm
  const float* x     = (const float*)d_in[0];
  const int*   tl    = (const int*)  d_in[2];
  const float* cmean = (const float*)d_in[4];
  const float* cstd  = (const float*)d_in[5];
  const float* wyom  = (const float*)d_in[6];
  const float* wylm  = (const float*)d_in[7];
  const float* wyfm  = (const float*)d_in[8];
  const float* b0o   = (const float*)d_in[9];
  const float* w1o   = (const float*)d_in[10];
  const float* w1l   = (const float*)d_in[11];
  const float* b0l   = (const float*)d_in[12];
  const float* w2o   = (const float*)d_in[13];
  const float* w2l   = (const float*)d_in[14];

  const int nsteps = in_sizes[0] / 2;    // B*T
  const int Bn     = nsteps / TSEQ;      // B

  mcpbrnn_scan<<<1, NTHREADS, 0, stream>>>(x, tl, cmean, cstd, wyom, wylm, wyfm,
                                           b0o, w1o, w1l, b0l, w2o, w2l,
                                           (float*)d_out, Bn, nsteps);
}